// ScanIntegrator_57629871178190
// MI455X (gfx1250) — compile-verified
//
#include <hip/hip_runtime.h>

// ---------------- problem constants (from reference) ----------------
#define NSTEPS 50
#define NXS    8
#define NUS    4

static constexpr float kDT    = 0.01f;
static constexpr float kDECAY = 0.5f;
static constexpr float kGAMMA = 0.26f;
static constexpr float kDTG   = kDT * kGAMMA;            // 0.0026
static constexpr float kC     = 1.0f + kDTG * kDECAY;    // Newton Jacobian diagonal
static constexpr float kINVC  = 1.0f / kC;
static constexpr float kKK    = kDTG / kC;               // Neumann contraction scale

typedef float v2f __attribute__((ext_vector_type(2)));
typedef float v8f __attribute__((ext_vector_type(8)));

// swap lane halves (0..15 <-> 16..31) within the wave.
// Preferred: v_permlanex16_b32 with identity selects -> dst[L] = src[L^16],
// pure VALU (short latency, no DS pipe / DScnt waits).
// Fallback: ds_swizzle SWAPX16.
static __device__ __forceinline__ float swapx16(float x) {
#if __has_builtin(__builtin_amdgcn_permlanex16)
  int xi = __float_as_int(x);
  int r  = __builtin_amdgcn_permlanex16(xi, xi, 0x76543210u, 0xFEDCBA98u, false, false);
  return __int_as_float(r);
#else
  return __int_as_float(__builtin_amdgcn_ds_swizzle(__float_as_int(x), 0x401F));
#endif
}

static __device__ __forceinline__ float fast_tanh(float x) {
#if __has_builtin(__builtin_amdgcn_tanhf)
  return __builtin_amdgcn_tanhf(x);
#else
  return tanhf(x);
#endif
}

// out = (v @ W1) for 32 batch elements held in canonical layout:
//   VGPR s = state s, lane L = element (groups a: lanes 0-15, b: lanes 16-31).
// Implemented as D = blockdiag(W1^T, W1^T) * B via 4x v_wmma_f32_16x16x4_f32.
// a[kc] holds the constant A tile chunk (A layout: M=lane%16, K=4kc+v+2*(lane/16)).
// B chunks packed from y[] with 8 half-swaps + 8 cndmasks; D comes back in the
// canonical layout, so no output shuffle is needed.
static __device__ __forceinline__ void matvec(const v2f a[4], const float y[8],
                                              bool lo, float out[8]) {
  float s0 = swapx16(y[0]), s1 = swapx16(y[1]);
  float s2 = swapx16(y[2]), s3 = swapx16(y[3]);
  float s4 = swapx16(y[4]), s5 = swapx16(y[5]);
  float s6 = swapx16(y[6]), s7 = swapx16(y[7]);
  v2f b0, b1, b2, b3;
  b0.x = lo ? y[0] : s2;  b0.y = lo ? y[1] : s3;   // K=0..3   (group a states 0..3)
  b1.x = lo ? y[4] : s6;  b1.y = lo ? y[5] : s7;   // K=4..7   (group a states 4..7)
  b2.x = lo ? s0 : y[2];  b2.y = lo ? s1 : y[3];   // K=8..11  (group b states 0..3)
  b3.x = lo ? s4 : y[6];  b3.y = lo ? s5 : y[7];   // K=12..15 (group b states 4..7)
  v8f c = {0.f, 0.f, 0.f, 0.f, 0.f, 0.f, 0.f, 0.f};
  c = __builtin_amdgcn_wmma_f32_16x16x4_f32(false, a[0], false, b0, (short)0, c, false, false);
  c = __builtin_amdgcn_wmma_f32_16x16x4_f32(false, a[1], false, b1, (short)0, c, false, false);
  c = __builtin_amdgcn_wmma_f32_16x16x4_f32(false, a[2], false, b2, (short)0, c, false, false);
  c = __builtin_amdgcn_wmma_f32_16x16x4_f32(false, a[3], false, b3, (short)0, c, false, false);
#pragma unroll
  for (int s = 0; s < 8; ++s) out[s] = c[s];
}

// f(y) = -DECAY*y + tanh(y@W1 + bias)
static __device__ __forceinline__ void feval(const v2f a[4], const float y[8],
                                             const float bias[8], bool lo, float f[8]) {
  float z[8];
  matvec(a, y, lo, z);
#pragma unroll
  for (int s = 0; s < 8; ++s) {
    float t = fast_tanh(z[s] + bias[s]);
    f[s] = t - kDECAY * y[s];
  }
}

// 4 fixed Newton iterations for y = cp + DTG*f(y).
// J^{-1} r via 2-term Neumann series (||M|| ~ 4e-3 -> trunc err ~1e-7, < f32 eps).
static __device__ __forceinline__ void newton4(const v2f a[4], const float cp[8],
                                               const float bias[8], bool lo, float y[8]) {
#pragma unroll 1
  for (int it = 0; it < 4; ++it) {
    float z[8];
    matvec(a, y, lo, z);
    float r[8], d[8];
#pragma unroll
    for (int s = 0; s < 8; ++s) {
      float t = fast_tanh(z[s] + bias[s]);
      float f = t - kDECAY * y[s];
      r[s] = y[s] - cp[s] - kDTG * f;   // residual G(y)
      d[s] = 1.0f - t * t;              // dsig
    }
    float q[8], p1[8];
    matvec(a, r, lo, q);
#pragma unroll
    for (int s = 0; s < 8; ++s) p1[s] = kKK * d[s] * q[s];   // M r
    matvec(a, p1, lo, q);
#pragma unroll
    for (int s = 0; s < 8; ++s) {
      float p2 = kKK * d[s] * q[s];                          // M^2 r
      y[s] -= kINVC * (r[s] + p1[s] + p2);
    }
  }
}

__global__ __launch_bounds__(256) void kvaerno5_scan_kernel(
    const float* __restrict__ X, const float* __restrict__ U,
    const float* __restrict__ W1, const float* __restrict__ W2,
    float* __restrict__ out) {
  const int  e    = blockIdx.x * blockDim.x + threadIdx.x;  // batch element
  const int  lane = threadIdx.x & 31;
  const bool lo   = lane < 16;
  const int  m    = lane & 15;
  const int  h    = lane >> 4;

  // Constant A tile: blockdiag(W1^T, W1^T) in WMMA f32 A-layout.
  // A(M,K) nonzero iff (M<8)==(K<8); value = W1[K&7][M&7] (i.e. W1^T).
  v2f a[4];
#pragma unroll
  for (int kc = 0; kc < 4; ++kc) {
    float av[2];
#pragma unroll
    for (int v = 0; v < 2; ++v) {
      int  K  = 4 * kc + v + 2 * h;
      bool on = ((m < 8) == (K < 8));
      av[v] = on ? W1[(K & 7) * NXS + (m & 7)] : 0.0f;
    }
    a[kc].x = av[0];
    a[kc].y = av[1];
  }

  // W2 is uniform across lanes: pin to SGPRs.
  float w2[NUS * NXS];
#pragma unroll
  for (int i = 0; i < NUS * NXS; ++i)
    w2[i] = __int_as_float(__builtin_amdgcn_readfirstlane(__float_as_int(W2[i])));

  const long xbase = (long)e * ((NSTEPS + 1) * NXS);
  const long ubase = (long)e * (NSTEPS * NUS);

  float y[8];
  {  // load y0, copy initial state to output row 0
    const float4* p  = (const float4*)(X + xbase);
    float4        r0 = p[0], r1 = p[1];
    y[0] = r0.x; y[1] = r0.y; y[2] = r0.z; y[3] = r0.w;
    y[4] = r1.x; y[5] = r1.y; y[6] = r1.z; y[7] = r1.w;
    float4* q = (float4*)(out + xbase);
    q[0] = r0;
    q[1] = r1;
  }

  // Kvaerno5 tableau rows (stages 2..7), zero-padded.
  constexpr float AR[6][6] = {
      {0.26f, 0.f, 0.f, 0.f, 0.f, 0.f},
      {0.13f, 0.84033320996790809f, 0.f, 0.f, 0.f, 0.f},
      {0.22371961478320505f, 0.47675532319799699f, -0.06470895363112615f, 0.f, 0.f, 0.f},
      {0.16648564323248321f, 0.10450018841591720f, 0.03631482272098715f,
       -0.13090704451073998f, 0.f, 0.f},
      {0.13855640231268224f, 0.0f, -0.04245337201752043f, 0.02446657898003141f,
       0.61943039072480676f, 0.f},
      {0.13659751177640291f, 0.0f, -0.05496908796538376f, -0.04118626728321046f,
       0.62993304899016403f, 0.06962479448202728f},
  };

#pragma unroll 1
  for (int step = 0; step < NSTEPS; ++step) {
    // per-step bias = clip(u) @ W2 (scalar-operand FMAs)
    float bias[8];
    {
      float4 uu  = *(const float4*)(U + ubase + step * NUS);
      float  uc0 = fminf(fmaxf(uu.x, -1.f), 1.f);
      float  uc1 = fminf(fmaxf(uu.y, -1.f), 1.f);
      float  uc2 = fminf(fmaxf(uu.z, -1.f), 1.f);
      float  uc3 = fminf(fmaxf(uu.w, -1.f), 1.f);
#pragma unroll
      for (int s = 0; s < 8; ++s)
        bias[s] = uc0 * w2[0 * NXS + s] + uc1 * w2[1 * NXS + s] +
                  uc2 * w2[2 * NXS + s] + uc3 * w2[3 * NXS + s];
    }

    float y0c[8], fs[6][8];
#pragma unroll
    for (int s = 0; s < 8; ++s) y0c[s] = y[s];

    feval(a, y, bias, lo, fs[0]);  // explicit first stage

#pragma unroll
    for (int r = 0; r < 6; ++r) {
      float cp[8];
#pragma unroll
      for (int s = 0; s < 8; ++s) {
        float acc = 0.f;
#pragma unroll
        for (int j = 0; j <= r; ++j) acc += AR[r][j] * fs[j][s];
        cp[s] = y0c[s] + kDT * acc;
      }
      newton4(a, cp, bias, lo, y);
      if (r < 5) feval(a, y, bias, lo, fs[r + 1]);  // last stage's f is never used
    }

    {  // store stage-7 value (b-row == last stage row)
      float4  o0 = {y[0], y[1], y[2], y[3]};
      float4  o1 = {y[4], y[5], y[6], y[7]};
      float4* q  = (float4*)(out + xbase + (long)(step + 1) * NXS);
      q[0] = o0;
      q[1] = o1;
    }
  }
}

extern "C" void kernel_launch(void* const* d_in, const int* in_sizes, int n_in,
                              void* d_out, int out_size, void* d_ws, size_t ws_size,
                              hipStream_t stream) {
  const float* X  = (const float*)d_in[0];
  const float* U  = (const float*)d_in[1];
  const float* W1 = (const float*)d_in[2];
  const float* W2 = (const float*)d_in[3];
  float*       out = (float*)d_out;
  const int    Btot = in_sizes[0] / ((NSTEPS + 1) * NXS);  // 65536
  dim3 block(256);                                          // 8 waves, 256 elements
  dim3 grid(Btot / 256);
  kvaerno5_scan_kernel<<<grid, block, 0, stream>>>(X, U, W1, W2, out);
}